// graph_embedding_Net_50594714746959
// MI455X (gfx1250) — compile-verified
//
#include <hip/hip_runtime.h>

// graph_embedding_Net for MI455X (gfx1250, wave32, WMMA f32 16x16x4)
// B=16, N=256, P=64, T=4 (fixed by reference constants)
// GEMM waves are register-blocked 16x64 (4 C tiles) to amortize A-fragment
// loads across 4 WMMAs per k-step.

#define B_SZ 16
#define N_SZ 256
#define P_SZ 64
#define T_IT 4
#define INV_N (1.0f / 256.0f)

typedef float v2f __attribute__((ext_vector_type(2)));
typedef float v8f __attribute__((ext_vector_type(8)));

// D = A(16x4,f32) * B(4x16,f32) + C(16x16,f32) on the matrix pipe.
__device__ __forceinline__ v8f wmma4(v8f c, v2f a, v2f b) {
  return __builtin_amdgcn_wmma_f32_16x16x4_f32(
      /*neg_a=*/false, a, /*neg_b=*/false, b,
      /*c_mod=*/(short)0, c, /*reuse_a=*/false, /*reuse_b=*/false);
}

// ---------------------------------------------------------------------------
// Stage 1: agg1[b,i,p]  = (1/n) * sum_j adjF[b,i,j]*relu(E[b,j,i]*W12[p]+b12[p])
//          agg34[b,i,p] = (1/n) * sum_j adjG[b,i,j]*relu(W[b,j,i]*W4[p] +b4[p])
// Non-GEMM (relu depends on (j,p), matrix depends on i) -> VALU.
// One 64-thread block per (b,i); E/W/adj loads are block-uniform broadcasts.
// ---------------------------------------------------------------------------
__global__ void agg_theta_kernel(const float* __restrict__ E,
                                 const float* __restrict__ W,
                                 const float* __restrict__ adjF,
                                 const float* __restrict__ adjG,
                                 const float* __restrict__ W12,
                                 const float* __restrict__ b12,
                                 const float* __restrict__ W4,
                                 const float* __restrict__ b4,
                                 float* __restrict__ agg1,
                                 float* __restrict__ agg34) {
  const int i = blockIdx.x;
  const int b = blockIdx.y;
  const int p = threadIdx.x;  // 0..63

  const float w12p = W12[p], b12p = b12[p];
  const float w4p  = W4[p],  b4p  = b4[p];

  const float* Ecol = E + (size_t)b * N_SZ * N_SZ + i;          // E[b][j][i]
  const float* Wcol = W + (size_t)b * N_SZ * N_SZ + i;          // W[b][j][i]
  const float* afr  = adjF + ((size_t)b * N_SZ + i) * N_SZ;     // adjF[b][i][j]
  const float* agr  = adjG + ((size_t)b * N_SZ + i) * N_SZ;     // adjG[b][i][j]

  float a1 = 0.0f, a2 = 0.0f;
#pragma unroll 4
  for (int j = 0; j < N_SZ; ++j) {
    const float ev = Ecol[(size_t)j * N_SZ];
    const float wv = Wcol[(size_t)j * N_SZ];
    a1 = fmaf(afr[j], fmaxf(fmaf(ev, w12p, b12p), 0.0f), a1);
    a2 = fmaf(agr[j], fmaxf(fmaf(wv, w4p,  b4p ), 0.0f), a2);
  }
  const size_t o = ((size_t)b * N_SZ + i) * P_SZ + p;
  agg1[o]  = a1 * INV_N;
  agg34[o] = a2 * INV_N;
}

// ---------------------------------------------------------------------------
// Stage 2: theta13 = agg1 @ W11.T + agg34 @ W3.T + (b11 + b3)
// Rows = B*N = 4096, cols = P = 64, K = P (twice).
// One wave -> 16x64 strip (4 C tiles). 256 waves -> 32 blocks of 8 waves.
// ---------------------------------------------------------------------------
__global__ void theta13_gemm_kernel(const float* __restrict__ agg1,
                                    const float* __restrict__ agg34,
                                    const float* __restrict__ W11,
                                    const float* __restrict__ b11,
                                    const float* __restrict__ W3,
                                    const float* __restrict__ b3,
                                    float* __restrict__ out) {
  const int lane = threadIdx.x & 31;
  const int wave = threadIdx.x >> 5;
  const int tid  = blockIdx.x * 8 + wave;        // 0..255 row strips
  const int row0 = tid * 16;
  const int l = lane & 15, g = lane >> 4, kb = g * 2;

  v8f acc[4] = {};
#pragma unroll 4
  for (int k0 = 0; k0 < P_SZ; k0 += 4) {
    v2f a;
    a.x = agg1[(size_t)(row0 + l) * P_SZ + k0 + kb];
    a.y = agg1[(size_t)(row0 + l) * P_SZ + k0 + kb + 1];
#pragma unroll
    for (int c = 0; c < 4; ++c) {
      v2f bm;
      bm.x = W11[(size_t)(c * 16 + l) * P_SZ + k0 + kb];   // (W11.T)[k][n]
      bm.y = W11[(size_t)(c * 16 + l) * P_SZ + k0 + kb + 1];
      acc[c] = wmma4(acc[c], a, bm);
    }
  }
#pragma unroll 4
  for (int k0 = 0; k0 < P_SZ; k0 += 4) {
    v2f a;
    a.x = agg34[(size_t)(row0 + l) * P_SZ + k0 + kb];
    a.y = agg34[(size_t)(row0 + l) * P_SZ + k0 + kb + 1];
#pragma unroll
    for (int c = 0; c < 4; ++c) {
      v2f bm;
      bm.x = W3[(size_t)(c * 16 + l) * P_SZ + k0 + kb];
      bm.y = W3[(size_t)(c * 16 + l) * P_SZ + k0 + kb + 1];
      acc[c] = wmma4(acc[c], a, bm);
    }
  }
#pragma unroll
  for (int c = 0; c < 4; ++c) {
    const float bias = b11[c * 16 + l] + b3[c * 16 + l];
#pragma unroll
    for (int r = 0; r < 8; ++r)
      out[(size_t)(row0 + r + 8 * g) * P_SZ + c * 16 + l] = acc[c][r] + bias;
  }
}

// ---------------------------------------------------------------------------
// Stage 3a (per iteration): t2 = MuIn @ W2.T + b2   ([4096,64]x[64,64])
// One wave -> 16x64 strip. 256 waves -> 32 blocks.
// ---------------------------------------------------------------------------
__global__ void t2_gemm_kernel(const float* __restrict__ MuIn,
                               const float* __restrict__ W2,
                               const float* __restrict__ b2,
                               float* __restrict__ t2) {
  const int lane = threadIdx.x & 31;
  const int wave = threadIdx.x >> 5;
  const int tid  = blockIdx.x * 8 + wave;        // 0..255
  const int row0 = tid * 16;
  const int l = lane & 15, g = lane >> 4, kb = g * 2;

  v8f acc[4] = {};
#pragma unroll 4
  for (int k0 = 0; k0 < P_SZ; k0 += 4) {
    v2f a;
    a.x = MuIn[(size_t)(row0 + l) * P_SZ + k0 + kb];
    a.y = MuIn[(size_t)(row0 + l) * P_SZ + k0 + kb + 1];
#pragma unroll
    for (int c = 0; c < 4; ++c) {
      v2f bm;
      bm.x = W2[(size_t)(c * 16 + l) * P_SZ + k0 + kb];    // (W2.T)[k][n]
      bm.y = W2[(size_t)(c * 16 + l) * P_SZ + k0 + kb + 1];
      acc[c] = wmma4(acc[c], a, bm);
    }
  }
#pragma unroll
  for (int c = 0; c < 4; ++c) {
    const float bias = b2[c * 16 + l];
#pragma unroll
    for (int r = 0; r < 8; ++r)
      t2[(size_t)(row0 + r + 8 * g) * P_SZ + c * 16 + l] = acc[c][r] + bias;
  }
}

// ---------------------------------------------------------------------------
// Stage 3b (per iteration): MuOut = relu((theta13 + (1/n)*adjG @ t2) / 3)
// Per-batch GEMM [256,256]x[256,64]; one wave -> 16x64 strip; fused epilogue.
// 16 batches x 16 strips = 256 waves -> 32 blocks.
// ---------------------------------------------------------------------------
__global__ void msg_gemm_kernel(const float* __restrict__ adjG,
                                const float* __restrict__ t2,
                                const float* __restrict__ theta13,
                                float* __restrict__ MuOut) {
  const int lane = threadIdx.x & 31;
  const int wave = threadIdx.x >> 5;
  const int tid  = blockIdx.x * 8 + wave;        // 0..255
  const int b    = tid >> 4;                     // 16 batches
  const int row0 = (tid & 15) * 16;              // 16 row strips
  const int l = lane & 15, g = lane >> 4, kb = g * 2;

  const float* A  = adjG + (size_t)b * N_SZ * N_SZ;
  const float* Bm = t2   + (size_t)b * N_SZ * P_SZ;

  v8f acc[4] = {};
#pragma unroll 4
  for (int k0 = 0; k0 < N_SZ; k0 += 4) {
    v2f a;
    a.x = A[(size_t)(row0 + l) * N_SZ + k0 + kb];
    a.y = A[(size_t)(row0 + l) * N_SZ + k0 + kb + 1];
#pragma unroll
    for (int c = 0; c < 4; ++c) {
      v2f bm;
      bm.x = Bm[(size_t)(k0 + kb) * P_SZ + c * 16 + l];
      bm.y = Bm[(size_t)(k0 + kb + 1) * P_SZ + c * 16 + l];
      acc[c] = wmma4(acc[c], a, bm);
    }
  }
  const size_t base = (size_t)b * N_SZ * P_SZ;
#pragma unroll
  for (int c = 0; c < 4; ++c) {
#pragma unroll
    for (int r = 0; r < 8; ++r) {
      const size_t o = base + (size_t)(row0 + r + 8 * g) * P_SZ + c * 16 + l;
      MuOut[o] = fmaxf((theta13[o] + acc[c][r] * INV_N) * (1.0f / 3.0f), 0.0f);
    }
  }
}

// ---------------------------------------------------------------------------
// Stage 4: readout. Tiny (16 graphs) -> single 64-thread block, VALU.
// ---------------------------------------------------------------------------
__global__ void readout_kernel(const float* __restrict__ Mu,
                               const int* __restrict__ e,
                               const float* __restrict__ W6,
                               const float* __restrict__ b6,
                               const float* __restrict__ W7,
                               const float* __restrict__ b7,
                               const float* __restrict__ W5,
                               const float* __restrict__ b5,
                               float* __restrict__ out) {
  __shared__ float musum[B_SZ][P_SZ];
  __shared__ float edgemu[B_SZ][P_SZ];
  __shared__ float th67[B_SZ][2 * P_SZ];

  const int p = threadIdx.x;  // 0..63
  for (int b = 0; b < B_SZ; ++b) {
    float acc = 0.0f;
    const float* Mb = Mu + (size_t)b * N_SZ * P_SZ;
#pragma unroll 4
    for (int i = 0; i < N_SZ; ++i) acc += Mb[(size_t)i * P_SZ + p];
    musum[b][p] = acc * INV_N;
    const int e0 = e[b * 2 + 0], e1 = e[b * 2 + 1];
    edgemu[b][p] = 0.5f * (Mb[(size_t)e0 * P_SZ + p] + Mb[(size_t)e1 * P_SZ + p]);
  }
  __syncthreads();

  const int q = threadIdx.x;  // output feature
  for (int b = 0; b < B_SZ; ++b) {
    float a6 = b6[q], a7 = b7[q];
#pragma unroll 8
    for (int pp = 0; pp < P_SZ; ++pp) {
      a6 = fmaf(musum[b][pp],  W6[(size_t)q * P_SZ + pp], a6);
      a7 = fmaf(edgemu[b][pp], W7[(size_t)q * P_SZ + pp], a7);
    }
    th67[b][q]        = fmaxf(a6, 0.0f);
    th67[b][P_SZ + q] = fmaxf(a7, 0.0f);
  }
  __syncthreads();

  if (threadIdx.x < 2 * B_SZ) {
    const int b = threadIdx.x >> 1, o = threadIdx.x & 1;
    float acc = b5[o];
#pragma unroll 8
    for (int c = 0; c < 2 * P_SZ; ++c)
      acc = fmaf(th67[b][c], W5[(size_t)o * 2 * P_SZ + c], acc);
    out[b * 2 + o] = acc;  // [b,1,2]
  }
}

// ---------------------------------------------------------------------------
extern "C" void kernel_launch(void* const* d_in, const int* in_sizes, int n_in,
                              void* d_out, int out_size, void* d_ws, size_t ws_size,
                              hipStream_t stream) {
  const float* E    = (const float*)d_in[0];
  const float* Mu   = (const float*)d_in[1];
  const float* W    = (const float*)d_in[2];
  const float* adjF = (const float*)d_in[3];
  const float* adjG = (const float*)d_in[4];
  const int*   e    = (const int*)d_in[5];
  // d_in[6..8] = T, n, bt_size (compile-time constants here)
  const float* W12 = (const float*)d_in[9];
  const float* b12 = (const float*)d_in[10];
  const float* W11 = (const float*)d_in[11];
  const float* b11 = (const float*)d_in[12];
  const float* W2  = (const float*)d_in[13];
  const float* b2  = (const float*)d_in[14];
  const float* W4  = (const float*)d_in[15];
  const float* b4  = (const float*)d_in[16];
  const float* W3  = (const float*)d_in[17];
  const float* b3  = (const float*)d_in[18];
  const float* W6  = (const float*)d_in[19];
  const float* b6  = (const float*)d_in[20];
  const float* W7  = (const float*)d_in[21];
  const float* b7  = (const float*)d_in[22];
  const float* W5  = (const float*)d_in[23];
  const float* b5  = (const float*)d_in[24];

  float* ws = (float*)d_ws;
  const size_t SZ = (size_t)B_SZ * N_SZ * P_SZ;  // 262144 floats (1 MB)
  float* theta13 = ws;            // persists entire run
  float* bufA    = ws + 1 * SZ;   // Mu ping
  float* bufB    = ws + 2 * SZ;   // Mu pong (aliases agg1 pre-loop: dead by t=1)
  float* t2buf   = ws + 3 * SZ;   // t2 (aliases agg34 pre-loop: dead by t=0)

  // Pre-stage: Mu-independent theta branches.
  agg_theta_kernel<<<dim3(N_SZ, B_SZ), P_SZ, 0, stream>>>(
      E, W, adjF, adjG, W12, b12, W4, b4, bufB /*agg1*/, t2buf /*agg34*/);
  theta13_gemm_kernel<<<32, 256, 0, stream>>>(
      bufB, t2buf, W11, b11, W3, b3, theta13);

  // T rounds of message passing (ping-pong, inputs never mutated).
  const float* cur = Mu;
  for (int t = 0; t < T_IT; ++t) {
    t2_gemm_kernel<<<32, 256, 0, stream>>>(cur, W2, b2, t2buf);
    float* nxt = (t & 1) ? bufB : bufA;
    msg_gemm_kernel<<<32, 256, 0, stream>>>(adjG, t2buf, theta13, nxt);
    cur = nxt;
  }

  // Readout (cur == bufB after T=4).
  readout_kernel<<<1, 64, 0, stream>>>(cur, e, W6, b6, W7, b7, W5, b5,
                                       (float*)d_out);
}